// HalfHadamardNoQuantizer_82798379532940
// MI455X (gfx1250) — compile-verified
//
#include <hip/hip_runtime.h>

typedef float v2f __attribute__((ext_vector_type(2)));
typedef float v8f __attribute__((ext_vector_type(8)));

#define C_DIM 4096
#define L_DIM 2048
#define BLK   64      // hadamard block size
#define LT    128     // l-span per workgroup
#define IS    144     // x-tile LDS row stride (dwords): 2*IS % 64 == 32 -> conflict-free A reads
#define HS    72      // H LDS row stride (dwords):      2*HS % 64 == 16 -> conflict-free B reads
#define OS    132     // out-tile LDS row stride (dwords), 16B-aligned rows

__global__ __launch_bounds__(256)
void hadamard64_wmma_f32(const float* __restrict__ x,
                         const float* __restrict__ H,
                         float* __restrict__ y) {
  // lds_xo: x-tile [64][IS] during compute (36864 B), reused as out-tile [64][OS] (33792 B)
  __shared__ float lds_xo[BLK * IS];
  __shared__ float lds_h[BLK * HS];

  const int tid = threadIdx.x;
  const int bi  = blockIdx.x;
  const int lb  = bi & 15;          // l-chunk:   L/LT  = 16
  const int g   = (bi >> 4) & 63;   // group:     C/BLK = 64
  const int b   = bi >> 10;         // batch
  const long base = (long)b * (C_DIM * (long)L_DIM)
                  + (long)(g * BLK) * L_DIM
                  + (long)(lb * LT);

  // ---- stage H: 64x64 f32 = 1024 float4, coalesced ----
  #pragma unroll
  for (int r = 0; r < 4; ++r) {
    int idx = r * 256 + tid;
    int row = idx >> 4;
    int c4  = (idx & 15) << 2;
    const float4 v = *(const float4*)(H + row * 64 + c4);
    *(float4*)(lds_h + row * HS + c4) = v;
  }
  // ---- stage x tile: [64 k][128 l] = 2048 float4, coalesced (512B per row) ----
  #pragma unroll
  for (int r = 0; r < 8; ++r) {
    int idx = r * 256 + tid;
    int row = idx >> 5;
    int c4  = (idx & 31) << 2;
    const float4 v = *(const float4*)(x + base + (long)row * L_DIM + c4);
    *(float4*)(lds_xo + row * IS + c4) = v;
  }
  __syncthreads();

  const int lane = tid & 31;
  const int wv   = tid >> 5;    // wave id -> l-tile (16 l's) within the 128-l chunk
  const int lr   = lane & 15;
  const int lh   = lane >> 4;   // half-wave selects K pair {k0,k0+1} vs {k0+2,k0+3}

  const int acol = wv * 16 + lr;      // l within chunk (A: lane = M = l)
  v8f acc[4] = {v8f{}, v8f{}, v8f{}, v8f{}};

  #pragma unroll
  for (int kc = 0; kc < 16; ++kc) {
    const int k0 = kc * 4 + lh * 2;
    v2f a;                                         // A 16x4 f32: v0=K even, v1=K odd
    a.x = lds_xo[k0 * IS + acol];
    a.y = lds_xo[(k0 + 1) * IS + acol];
    #pragma unroll
    for (int jt = 0; jt < 4; ++jt) {
      v2f bf;                                      // B 4x16 f32: lane = N = j, mirrored K layout
      bf.x = lds_h[k0 * HS + jt * 16 + lr];
      bf.y = lds_h[(k0 + 1) * HS + jt * 16 + lr];
      acc[jt] = __builtin_amdgcn_wmma_f32_16x16x4_f32(
          false, a, false, bf, (short)0, acc[jt], false, false);
    }
  }
  __syncthreads();   // all waves done reading lds_xo before it is reused as out-tile

  // ---- transpose accumulators through LDS: D lane = j, VGPR = l-row ----
  #pragma unroll
  for (int jt = 0; jt < 4; ++jt) {
    const int j = jt * 16 + lr;
    #pragma unroll
    for (int i = 0; i < 8; ++i) {
      const int m = wv * 16 + lh * 8 + i;          // l within chunk
      lds_xo[j * OS + m] = acc[jt][i];
    }
  }
  __syncthreads();

  // ---- coalesced store: rows j contiguous along l (512B per row) ----
  #pragma unroll
  for (int r = 0; r < 8; ++r) {
    int idx = r * 256 + tid;
    int row = idx >> 5;
    int c4  = (idx & 31) << 2;
    const float4 v = *(const float4*)(lds_xo + row * OS + c4);
    *(float4*)(y + base + (long)row * L_DIM + c4) = v;
  }
}

extern "C" void kernel_launch(void* const* d_in, const int* in_sizes, int n_in,
                              void* d_out, int out_size, void* d_ws, size_t ws_size,
                              hipStream_t stream) {
  const float* x = (const float*)d_in[0];
  const float* H = (const float*)d_in[1];
  float* y = (float*)d_out;

  const int total = in_sizes[0];                 // B * C * L
  const int Bn = total / (C_DIM * L_DIM);        // 8
  const int nblocks = Bn * (C_DIM / BLK) * (L_DIM / LT);   // 8 * 64 * 16 = 8192
  hadamard64_wmma_f32<<<nblocks, 256, 0, stream>>>(x, H, y);
}